// VectorQuantizer_37383395344485
// MI455X (gfx1250) — compile-verified
//
#include <hip/hip_runtime.h>
#include <math.h>

typedef __attribute__((ext_vector_type(2))) float v2f;
typedef __attribute__((ext_vector_type(8))) float v8f;

#define N_E    1024
#define E_DIM  64
#define N_ROWS 32768            // 32*32*32 rows of z_flat
#define HW     1024             // 32*32
#define BETA   0.25f
#define N_ZQ   2097152          // 32*64*32*32

// d_out element offsets (floats), concat of (loss, z_q, perplexity, one_hot, indices)
#define O_LOSS 0
#define O_ZQ   1
#define O_PERP 2097153
#define O_ENC  2097154
#define O_IDX  35651586

// workspace byte offsets
#define WS_ENORM 0              // 1024 f32
#define WS_IDX   4096           // 32768 i32
#define WS_HIST  135168         // 1024 i32
#define WS_PART  139264         // 8192 f32
#define NPART    8192

// LDS codebook staging
#define CHUNK 128               // codes per LDS chunk (8 chunks total)
#define PITCH 68                // padded floats per code row: bank-conflict-free

// ---------------------------------------------------------------------------
// 1) per-code squared norms + zero histogram
__global__ void vq_prep(const float* __restrict__ emb,
                        float* __restrict__ enorm, int* __restrict__ hist) {
    int n = blockIdx.x * blockDim.x + threadIdx.x;   // 0..1023
    const float* e = emb + n * E_DIM;
    float s = 0.f;
    #pragma unroll
    for (int k = 0; k < E_DIM; ++k) s += e[k] * e[k];
    enorm[n] = s;
    hist[n]  = 0;
}

// ---------------------------------------------------------------------------
// 2) WMMA distance GEMM + fused argmin. One wave per 16-row M tile.
//    Codebook staged through LDS in 128-code chunks shared by all 8 waves.
__launch_bounds__(256)
__global__ void vq_argmin_wmma(const float* __restrict__ z,
                               const float* __restrict__ emb,
                               const float* __restrict__ enorm,
                               int* __restrict__ out_idx) {
    __shared__ float sB[CHUNK * PITCH];   // 34,816 B padded codebook chunk
    __shared__ float sN[CHUNK];           // chunk code norms

    const int lane = threadIdx.x;                 // 0..31
    const int wave = threadIdx.y;                 // 0..7
    const int tid  = wave * 32 + lane;            // 0..255
    const int tile = blockIdx.x * 8 + wave;       // 0..2047
    const int r    = lane & 15;
    const int kb   = (lane >> 4) * 2;             // K half: 0 or 2

    // row n = tile*16 + r; tiles never straddle a batch (16 | 1024)
    const int n0 = tile * 16;
    const int b  = n0 >> 10;
    const int hw = (n0 & 1023) + r;
    const float* zrow = z + (size_t)b * (E_DIM * HW) + hw;  // elem c at zrow[c*HW]

    // A tile resident in registers: 16 chunks of 16x4 fp32
    // (lane r = row M, VGPR0/1 = K kb/kb+1 per ISA A-layout)
    v2f a[16];
    #pragma unroll
    for (int kc = 0; kc < 16; ++kc) {
        int k0 = kc * 4 + kb;
        a[kc].x = zrow[(size_t)k0 * HW];
        a[kc].y = zrow[(size_t)(k0 + 1) * HW];
    }

    float bestv[8];
    int   besti[8];
    #pragma unroll
    for (int v = 0; v < 8; ++v) { bestv[v] = 3.4e38f; besti[v] = 0; }

    const int code_cp = tid >> 1;   // code this thread helps stage (0..127)
    const int half_cp = tid & 1;    // which 32-float half of the 64-dim row

    for (int ch = 0; ch < N_E / CHUNK; ++ch) {
        __syncthreads();
        // cooperative stage: 128 codes x 64 f32 -> padded LDS (float4 copies)
        const float4* src =
            (const float4*)(emb + ((size_t)ch * CHUNK + code_cp) * E_DIM) + half_cp * 8;
        float4* dst = (float4*)(sB + code_cp * PITCH) + half_cp * 8;
        #pragma unroll
        for (int i = 0; i < 8; ++i) dst[i] = src[i];
        if (tid < CHUNK) sN[tid] = enorm[ch * CHUNK + tid];
        __syncthreads();

        for (int t = 0; t < CHUNK / 16; ++t) {    // 8 N-tiles per chunk
            const int ncl  = t * 16 + r;          // code within chunk (lane's column)
            const float* erow = sB + ncl * PITCH; // conflict-free: banks 4r(+2)+4kc

            // preload full K=64 B-column into registers so the DS loads issue
            // as one clause instead of a wait before every WMMA pair
            v2f bbv[16];
            #pragma unroll
            for (int kc = 0; kc < 16; ++kc) {
                int k0 = kc * 4 + kb;
                bbv[kc].x = erow[k0];
                bbv[kc].y = erow[k0 + 1];
            }

            v8f c = {};
            #pragma unroll
            for (int kc = 0; kc < 16; ++kc) {
                c = __builtin_amdgcn_wmma_f32_16x16x4_f32(
                        false, a[kc], false, bbv[kc], (short)0, c, false, false);
            }

            const float en   = sN[ncl];
            const int ncode  = ch * CHUNK + ncl;
            #pragma unroll
            for (int v = 0; v < 8; ++v) {
                float d = en - 2.0f * c[v];       // ||z||^2 constant per row: dropped
                if (d < bestv[v]) { bestv[v] = d; besti[v] = ncode; }  // keeps lowest idx
            }
        }
    }

    // min-reduce across the 16 lanes of each half (N residues), lower-idx tie-break
    #pragma unroll
    for (int m = 1; m < 16; m <<= 1) {
        #pragma unroll
        for (int v = 0; v < 8; ++v) {
            float ov = __shfl_xor(bestv[v], m, 32);
            int   oi = __shfl_xor(besti[v], m, 32);
            if (ov < bestv[v] || (ov == bestv[v] && oi < besti[v])) {
                bestv[v] = ov; besti[v] = oi;
            }
        }
    }
    if (lane == 0) {            // rows 0..7 of tile
        #pragma unroll
        for (int v = 0; v < 8; ++v) out_idx[n0 + v] = besti[v];
    } else if (lane == 16) {    // rows 8..15 of tile
        #pragma unroll
        for (int v = 0; v < 8; ++v) out_idx[n0 + 8 + v] = besti[v];
    }
}

// ---------------------------------------------------------------------------
// 3) indices -> float output + integer histogram
__global__ void vq_hist(const int* __restrict__ idx,
                        float* __restrict__ out_idx_f, int* __restrict__ hist) {
    int n = blockIdx.x * blockDim.x + threadIdx.x;   // 0..32767
    int k = idx[n];
    out_idx_f[n] = (float)k;
    atomicAdd(&hist[k], 1);
}

// ---------------------------------------------------------------------------
// 4) z_q (B,C,H,W) write + fused per-block sum of (z_q - z_p)^2
__launch_bounds__(256)
__global__ void vq_zq_loss(const float* __restrict__ z, const float* __restrict__ emb,
                           const int* __restrict__ idx, float* __restrict__ zq_out,
                           float* __restrict__ partial) {
    int gid = blockIdx.x * 256 + threadIdx.x;   // layout: b*65536 + c*1024 + hw
    int b   = gid >> 16;
    int c   = (gid >> 10) & 63;
    int hw  = gid & 1023;
    int n   = b * 1024 + hw;
    float zq = emb[idx[n] * E_DIM + c];
    float zp = z[gid];
    zq_out[gid] = zq;                            // straight-through value == z_q
    float d = zq - zp;
    d = d * d;

    __shared__ float s[256];
    s[threadIdx.x] = d;
    __syncthreads();
    for (int off = 128; off > 0; off >>= 1) {
        if (threadIdx.x < off) s[threadIdx.x] += s[threadIdx.x + off];
        __syncthreads();
    }
    if (threadIdx.x == 0) partial[blockIdx.x] = s[0];
}

// ---------------------------------------------------------------------------
// 5) one-hot encodings: one block per row, float2 stores (section is 8B aligned)
__launch_bounds__(256)
__global__ void vq_onehot(const int* __restrict__ idx, float* __restrict__ enc) {
    int row  = blockIdx.x;
    int best = idx[row];
    float* rp = enc + (size_t)row * N_E;
    int c0 = threadIdx.x * 4;
    float2 w0 = make_float2(c0 == best ? 1.f : 0.f, c0 + 1 == best ? 1.f : 0.f);
    float2 w1 = make_float2(c0 + 2 == best ? 1.f : 0.f, c0 + 3 == best ? 1.f : 0.f);
    *(float2*)(rp + c0)     = w0;
    *(float2*)(rp + c0 + 2) = w1;
}

// ---------------------------------------------------------------------------
// 6) final deterministic reduction: loss + perplexity
__launch_bounds__(1024)
__global__ void vq_final(const float* __restrict__ partial, const int* __restrict__ hist,
                         float* __restrict__ loss_out, float* __restrict__ perp_out) {
    __shared__ float sl[1024];
    __shared__ float sp[1024];
    int t = threadIdx.x;
    float s = 0.f;
    for (int i = t; i < NPART; i += 1024) s += partial[i];
    sl[t] = s;
    float e = (float)hist[t] * (1.0f / (float)N_ROWS);
    sp[t] = e * logf(e + 1e-10f);
    __syncthreads();
    for (int off = 512; off > 0; off >>= 1) {
        if (t < off) { sl[t] += sl[t + off]; sp[t] += sp[t + off]; }
        __syncthreads();
    }
    if (t == 0) {
        loss_out[0] = (1.0f + BETA) * sl[0] * (1.0f / (float)N_ZQ);
        perp_out[0] = expf(-sp[0]);
    }
}

// ---------------------------------------------------------------------------
extern "C" void kernel_launch(void* const* d_in, const int* in_sizes, int n_in,
                              void* d_out, int out_size, void* d_ws, size_t ws_size,
                              hipStream_t stream) {
    const float* z   = (const float*)d_in[0];
    const float* emb = (const float*)d_in[1];
    float* out = (float*)d_out;
    char*  ws  = (char*)d_ws;

    float* enorm   = (float*)(ws + WS_ENORM);
    int*   idx     = (int*)(ws + WS_IDX);
    int*   hist    = (int*)(ws + WS_HIST);
    float* partial = (float*)(ws + WS_PART);

    vq_prep       <<<N_E / 256, 256, 0, stream>>>(emb, enorm, hist);
    vq_argmin_wmma<<<256, dim3(32, 8), 0, stream>>>(z, emb, enorm, idx);
    vq_hist       <<<N_ROWS / 256, 256, 0, stream>>>(idx, out + O_IDX, hist);
    vq_zq_loss    <<<N_ZQ / 256, 256, 0, stream>>>(z, emb, idx, out + O_ZQ, partial);
    vq_onehot     <<<N_ROWS, 256, 0, stream>>>(idx, out + O_ENC);
    vq_final      <<<1, 1024, 0, stream>>>(partial, hist, out + O_LOSS, out + O_PERP);
}